// Solver_73237782331777
// MI455X (gfx1250) — compile-verified
//
#include <hip/hip_runtime.h>
#include <hip/hip_bf16.h>
#include <cstdint>
#include <cstddef>

#define BLOCK     256
#define HALO_W    (BLOCK + 2)          // 258 columns
#define LDS_WORDS (3 * HALO_W)         // 3 rows  -> 774 floats = 3096 B
#define MAXVAL    1000.0f
#define EPSVAL    1e-12f
#define NUM_ITERS 100

struct F3 { float p, q, r; };          // 12 bytes -> b96 load/store

// ---------------------------------------------------------------------------
// CDNA5 async global -> LDS copy (per-lane LDS destination), guarded.
// Builtin signature (probe-confirmed): (AS1 int*, AS3 int*, imm, imm)
// ---------------------------------------------------------------------------
__device__ __forceinline__ void async_u32(const float* gsrc, float* ldst) {
#if defined(__gfx1250__) && __has_builtin(__builtin_amdgcn_global_load_async_to_lds_b32)
  typedef __attribute__((address_space(1))) int gint_t;
  typedef __attribute__((address_space(3))) int lint_t;
  __builtin_amdgcn_global_load_async_to_lds_b32(
      (gint_t*)gsrc, (lint_t*)ldst, 0, 0);   // proper addrspacecast, not int reinterpret
#else
  *ldst = *gsrc;
#endif
}

__device__ __forceinline__ void wait_async_all() {
#if defined(__gfx1250__)
#if __has_builtin(__builtin_amdgcn_s_wait_asynccnt)
  __builtin_amdgcn_s_wait_asynccnt(0);
#else
  asm volatile("s_wait_asynccnt 0" ::: "memory");
#endif
#endif
}

// ---------------------------------------------------------------------------
// Load the 3 x (BLOCK+2) halo of u around this block's half-row into LDS.
// ---------------------------------------------------------------------------
__device__ __forceinline__ void load_halo(const float* __restrict__ uIn, float* tile,
                                          int rm1, int c0m1, int n) {
  for (int idx = threadIdx.x; idx < LDS_WORDS; idx += BLOCK) {
    int dr = idx / HALO_W;           // constant divisor -> mul
    int dc = idx - dr * HALO_W;
    int gr = rm1 + dr;  gr = gr < 0 ? 0 : (gr >= n ? n - 1 : gr);
    int gc = c0m1 + dc; gc = gc < 0 ? 0 : (gc >= n ? n - 1 : gc);
    async_u32(&uIn[gr * n + gc], &tile[idx]);
  }
  wait_async_all();
  __syncthreads();
}

// Fetch u[idx]: LDS tile if inside halo, else global (rare / boundary-safe).
__device__ __forceinline__ float load_u(const float* __restrict__ uIn, const float* tile,
                                        int idx, int rm1, int c0m1, int n, int nShift,
                                        int useTile) {
  if (useTile) {
    int jr = (nShift >= 0) ? (idx >> nShift) : (idx / n);
    int jc = idx - jr * n;
    int dr = jr - rm1;
    int dc = jc - c0m1;
    if (((unsigned)dr < 3u) && ((unsigned)dc < (unsigned)HALO_W))
      return tile[dr * HALO_W + dc];
  }
  return uIn[idx];
}

// ---------------------------------------------------------------------------
// Eikonal candidate evaluation (matches reference _global_update math).
// ---------------------------------------------------------------------------
__device__ __forceinline__ float eikonal_min(float p, float q, float r,
                                             float Tj, float Tk, float best) {
  float dT = Tj - Tk;
  float qs = (q > EPSVAL) ? q : 1.0f;
  float denom = q - dT * dT;
  float denomSafe = (fabsf(denom) > EPSVAL) ? denom : 1.0f;
  float rad = (q * r - p * p) / denomSafe;
  bool root_ok = (denom > EPSVAL) && (rad >= 0.0f);
  float sq = sqrtf(fmaxf(rad, 0.0f));
  float lam1 = (p - dT * sq) / qs;
  float lam2 = (p + dT * sq) / qs;
  // travel(0), travel(1)
  float c0 = Tk + sqrtf(fmaxf(r, EPSVAL));
  float c1 = Tk + dT + sqrtf(fmaxf(r - 2.0f * p + q, EPSVAL));
  best = fminf(best, fminf(c0, c1));
  if (root_ok) {
    if (lam1 > 0.0f && lam1 < 1.0f) {
      float t = Tk + lam1 * dT + sqrtf(fmaxf(r - 2.0f * lam1 * p + lam1 * lam1 * q, EPSVAL));
      best = fminf(best, t);
    }
    if (lam2 > 0.0f && lam2 < 1.0f) {
      float t = Tk + lam2 * dT + sqrtf(fmaxf(r - 2.0f * lam2 * p + lam2 * lam2 * q, EPSVAL));
      best = fminf(best, t);
    }
  }
  return best;
}

// ---------------------------------------------------------------------------
// u0 = MAXVAL; u0[initial_inds] = initial_vals
// ---------------------------------------------------------------------------
__global__ __launch_bounds__(BLOCK) void init_kernel(float* __restrict__ u, int Nv,
                                                     const int* __restrict__ inds,
                                                     const float* __restrict__ vals,
                                                     int nInit) {
  int v = blockIdx.x * BLOCK + threadIdx.x;
  if (v >= Nv) return;
  float val = MAXVAL;
  for (int i = 0; i < nInit; ++i)
    if (inds[i] == v) val = vals[i];
  u[v] = val;
}

// ---------------------------------------------------------------------------
// One-time geometry precompute: {p,q,r} and {j,k} per (entry e, vertex v),
// stored entry-major (idx = e*Nv + v) for coalesced streaming each sweep.
// Invalid entries flagged with j = -1.
// ---------------------------------------------------------------------------
__global__ __launch_bounds__(BLOCK) void precompute_kernel(
    const float* __restrict__ Mf, const float* __restrict__ verts,
    const int* __restrict__ adj,
    F3* __restrict__ GEO, int2* __restrict__ JK,
    int Nv, int K) {
  long t = (long)blockIdx.x * BLOCK + threadIdx.x;
  long total = (long)Nv * K;
  if (t >= total) return;
  int v = (int)(t % Nv);
  int e = (int)(t / Nv);
  int4 a4 = ((const int4*)adj)[(size_t)v * K + e];   // {ci, j, k, s}
  int j = a4.y, k = a4.z, s = a4.w;
  if (s < 0) {
    GEO[t] = F3{0.0f, 0.0f, 0.0f};
    JK[t] = make_int2(-1, 0);
    return;
  }
  float2 xi = ((const float2*)verts)[v];
  float2 xj = ((const float2*)verts)[j];
  float2 xk = ((const float2*)verts)[k];
  float ax = xi.x - xk.x, ay = xi.y - xk.y;
  float bx = xj.x - xk.x, by = xj.y - xk.y;
  float4 m = ((const float4*)Mf)[s];                 // M00 M01 M10 M11
  float Max = m.x * ax + m.y * ay;
  float May = m.z * ax + m.w * ay;
  float Mbx = m.x * bx + m.y * by;
  float Mby = m.z * bx + m.w * by;
  F3 g;
  g.p = bx * Max + by * May;   // p = b.Ma
  g.q = bx * Mbx + by * Mby;   // q = b.Mb
  g.r = ax * Max + ay * May;   // r = a.Ma
  GEO[t] = g;
  JK[t] = make_int2(j, k);
}

// ---------------------------------------------------------------------------
// Jacobi sweep using precomputed geometry (primary path).
// KT > 0: compile-time entry count (fully unrolled); KT == 0: runtime K.
// ---------------------------------------------------------------------------
template <int KT>
__global__ __launch_bounds__(BLOCK) void iter_pre_kernel(
    const float* __restrict__ uIn, float* __restrict__ uOut,
    const F3* __restrict__ GEO, const int2* __restrict__ JK,
    int Nv, int Krt, int n, int nShift, int useTile,
    const int* __restrict__ inds, const float* __restrict__ vals, int nInit) {
  extern __shared__ float tile[];
  const int v = blockIdx.x * BLOCK + threadIdx.x;
  int rm1 = 0, c0m1 = 0;
  if (useTile) {
    int v0 = blockIdx.x * BLOCK;
    int r  = (nShift >= 0) ? (v0 >> nShift) : (v0 / n);
    int c0 = v0 - r * n;
    rm1 = r - 1; c0m1 = c0 - 1;
    load_halo(uIn, tile, rm1, c0m1, n);
  }
  if (v >= Nv) return;
  const int K = (KT > 0) ? KT : Krt;
  float best = load_u(uIn, tile, v, rm1, c0m1, n, nShift, useTile);
#pragma unroll
  for (int e = 0; e < K; ++e) {
    long base = (long)e * Nv + v;
    int2 jk = JK[base];      // b64
    F3 g = GEO[base];        // b96
    if (jk.x >= 0) {
      float Tj = load_u(uIn, tile, jk.x, rm1, c0m1, n, nShift, useTile);
      float Tk = load_u(uIn, tile, jk.y, rm1, c0m1, n, nShift, useTile);
      best = eikonal_min(g.p, g.q, g.r, Tj, Tk, best);
    }
  }
  for (int i = 0; i < nInit; ++i)
    if (inds[i] == v) best = vals[i];
  uOut[v] = best;
}

// ---------------------------------------------------------------------------
// Fallback sweep: recompute geometry on the fly (if d_ws too small).
// ---------------------------------------------------------------------------
__global__ __launch_bounds__(BLOCK) void iter_fly_kernel(
    const float* __restrict__ uIn, float* __restrict__ uOut,
    const float* __restrict__ Mf, const float* __restrict__ verts,
    const int* __restrict__ adj,
    int Nv, int K, int n, int nShift, int useTile,
    const int* __restrict__ inds, const float* __restrict__ vals, int nInit) {
  extern __shared__ float tile[];
  const int v = blockIdx.x * BLOCK + threadIdx.x;
  int rm1 = 0, c0m1 = 0;
  if (useTile) {
    int v0 = blockIdx.x * BLOCK;
    int r  = (nShift >= 0) ? (v0 >> nShift) : (v0 / n);
    int c0 = v0 - r * n;
    rm1 = r - 1; c0m1 = c0 - 1;
    load_halo(uIn, tile, rm1, c0m1, n);
  }
  if (v >= Nv) return;
  float best = load_u(uIn, tile, v, rm1, c0m1, n, nShift, useTile);
  float2 xi = ((const float2*)verts)[v];
  for (int e = 0; e < K; ++e) {
    int4 a4 = ((const int4*)adj)[(size_t)v * K + e];
    int j = a4.y, k = a4.z, s = a4.w;
    if (s >= 0) {
      float2 xj = ((const float2*)verts)[j];
      float2 xk = ((const float2*)verts)[k];
      float ax = xi.x - xk.x, ay = xi.y - xk.y;
      float bx = xj.x - xk.x, by = xj.y - xk.y;
      float4 m = ((const float4*)Mf)[s];
      float Max = m.x * ax + m.y * ay;
      float May = m.z * ax + m.w * ay;
      float Mbx = m.x * bx + m.y * by;
      float Mby = m.z * bx + m.w * by;
      float p = bx * Max + by * May;
      float q = bx * Mbx + by * Mby;
      float r = ax * Max + ay * May;
      float Tj = load_u(uIn, tile, j, rm1, c0m1, n, nShift, useTile);
      float Tk = load_u(uIn, tile, k, rm1, c0m1, n, nShift, useTile);
      best = eikonal_min(p, q, r, Tj, Tk, best);
    }
  }
  for (int i = 0; i < nInit; ++i)
    if (inds[i] == v) best = vals[i];
  uOut[v] = best;
}

// ---------------------------------------------------------------------------
// Host driver
// ---------------------------------------------------------------------------
extern "C" void kernel_launch(void* const* d_in, const int* in_sizes, int n_in,
                              void* d_out, int out_size, void* d_ws, size_t ws_size,
                              hipStream_t stream) {
  const float* Mf    = (const float*)d_in[0];   // (S,2,2)
  const float* verts = (const float*)d_in[1];   // (Nv,2)
  const int*   adj   = (const int*)d_in[2];     // (Nv,K,4)
  const int*   inds  = (const int*)d_in[3];
  const float* vals  = (const float*)d_in[4];

  const int Nv    = in_sizes[1] / 2;
  const int K     = in_sizes[2] / (Nv * 4);
  const int nInit = in_sizes[3];

  // grid width (n x n vertices)
  int n = 1;
  while ((long)n * n < (long)Nv) ++n;
  const int isSquare = ((long)n * n == (long)Nv);
  int nShift = -1;
  if ((n & (n - 1)) == 0) { nShift = 0; while ((1 << nShift) < n) ++nShift; }
  const int useTile = isSquare && (n % BLOCK == 0);

  // workspace layout: [u ping buffer][GEO (12B/entry)][JK (8B/entry)]
  float* uW = (ws_size >= (size_t)Nv * 4) ? (float*)d_ws : (float*)d_out;
  size_t off0 = (((size_t)Nv * 4) + 255) & ~(size_t)255;
  size_t nEnt = (size_t)Nv * K;
  size_t geoBytes = ((nEnt * sizeof(F3)) + 15) & ~(size_t)15;
  int usePre = (ws_size >= off0 + geoBytes + nEnt * sizeof(int2)) ? 1 : 0;
  F3*   GEO = (F3*)((char*)d_ws + off0);
  int2* JK  = (int2*)((char*)d_ws + off0 + geoBytes);

  const int gridV = (Nv + BLOCK - 1) / BLOCK;
  const size_t shBytes = LDS_WORDS * sizeof(float);

  // u0 lives in d_out so that after an even number of sweeps the result
  // is back in d_out (no final copy needed).
  float* uO = (float*)d_out;
  init_kernel<<<gridV, BLOCK, 0, stream>>>(uO, Nv, inds, vals, nInit);

  if (usePre) {
    long total = (long)Nv * K;
    int gridE = (int)((total + BLOCK - 1) / BLOCK);
    precompute_kernel<<<gridE, BLOCK, 0, stream>>>(Mf, verts, adj, GEO, JK, Nv, K);
  }

  const float* cur = uO;
  for (int it = 0; it < NUM_ITERS; ++it) {
    float* dst = (cur == uO) ? uW : uO;
    if (usePre) {
      if (K == 6) {
        iter_pre_kernel<6><<<gridV, BLOCK, shBytes, stream>>>(
            cur, dst, GEO, JK, Nv, K, n, nShift, useTile, inds, vals, nInit);
      } else {
        iter_pre_kernel<0><<<gridV, BLOCK, shBytes, stream>>>(
            cur, dst, GEO, JK, Nv, K, n, nShift, useTile, inds, vals, nInit);
      }
    } else {
      iter_fly_kernel<<<gridV, BLOCK, shBytes, stream>>>(
          cur, dst, Mf, verts, adj, Nv, K, n, nShift, useTile, inds, vals, nInit);
    }
    cur = dst;
  }
  // NUM_ITERS is even -> cur == uO == d_out
}